// EmbeddingLayer_21749714386979
// MI455X (gfx1250) — compile-verified
//
#include <hip/hip_runtime.h>
#include <hip/hip_bf16.h>
#include <stdint.h>

// ---------------------------------------------------------------------------
// Embedding gather: out[t, :] = W[x[t], :]   (B*S = 16384 tokens, D = 1024 f32)
//
// Pure data movement (0 FLOPs, ~128 MiB traffic -> ~5.6us at 23.3 TB/s).
// Implemented with the CDNA5 Tensor Data Mover in GATHER mode:
//   - one TENSOR_LOAD_TO_LDS per workgroup pulls 8 indexed 4KB rows of W
//     into LDS (row indices embedded in D# groups 2/3, 32-bit index mode)
//   - one TENSOR_STORE_FROM_LDS streams the 32KB tile contiguously to out
// Tracked by TENSORcnt; no VGPR staging, no per-element address math.
// ---------------------------------------------------------------------------

typedef __attribute__((ext_vector_type(4))) unsigned int u32x4;
typedef __attribute__((ext_vector_type(4))) int          i32x4;
typedef __attribute__((ext_vector_type(8))) int          i32x8;

#define EMB_D        1024u     // d_model (floats per row)
#define EMB_VOCAB    50257u
#define ROWS_PER_WG  8u        // 8 x 32-bit gather indices per descriptor

// ROCm 7.2 (clang-22): 5-arg builtin.  amdgpu-toolchain (clang-23): 6-arg.
#if defined(__clang_major__) && (__clang_major__ >= 23)
#define TDM_LOAD(g0, g1, g2, g3) \
  __builtin_amdgcn_tensor_load_to_lds((g0), (g1), (g2), (g3), (i32x8){0,0,0,0,0,0,0,0}, 0)
#define TDM_STORE(g0, g1, g2, g3) \
  __builtin_amdgcn_tensor_store_from_lds((g0), (g1), (g2), (g3), (i32x8){0,0,0,0,0,0,0,0}, 0)
#else
#define TDM_LOAD(g0, g1, g2, g3) \
  __builtin_amdgcn_tensor_load_to_lds((g0), (g1), (g2), (g3), 0)
#define TDM_STORE(g0, g1, g2, g3) \
  __builtin_amdgcn_tensor_store_from_lds((g0), (g1), (g2), (g3), 0)
#endif

__global__ __launch_bounds__(32) void embed_gather_tdm(
    const int* __restrict__ x,      // (B*S) token ids
    const float* __restrict__ W,    // (VOCAB, D) table
    float* __restrict__ out)        // (B*S, D)
{
  __shared__ float tile[ROWS_PER_WG * EMB_D];   // 32 KB LDS staging buffer

  const uint32_t wg      = blockIdx.x;
  const uint32_t rowBase = wg * ROWS_PER_WG;

  // Gather indices -> SGPRs (TDM descriptors are SGPR groups).
  const int i0 = __builtin_amdgcn_readfirstlane(x[rowBase + 0]);
  const int i1 = __builtin_amdgcn_readfirstlane(x[rowBase + 1]);
  const int i2 = __builtin_amdgcn_readfirstlane(x[rowBase + 2]);
  const int i3 = __builtin_amdgcn_readfirstlane(x[rowBase + 3]);
  const int i4 = __builtin_amdgcn_readfirstlane(x[rowBase + 4]);
  const int i5 = __builtin_amdgcn_readfirstlane(x[rowBase + 5]);
  const int i6 = __builtin_amdgcn_readfirstlane(x[rowBase + 6]);
  const int i7 = __builtin_amdgcn_readfirstlane(x[rowBase + 7]);

  // Flat 'shared' pointers carry the LDS byte offset in their low 32 bits.
  const uint32_t lds_off = (uint32_t)(uintptr_t)&tile[0];

  const uint64_t wAddr = (uint64_t)(uintptr_t)W;
  const uint64_t oAddr = (uint64_t)(uintptr_t)(out + (size_t)rowBase * EMB_D);

  // ---------------- load descriptor: gather mode, 32-bit indices ----------
  // group0: count=1 | gather_index_size(32b)=1<<30 | gather_mode=1<<31;
  //         lds_addr; global_addr[56:0]; type=2 ("image")
  u32x4 lg0;
  lg0[0] = 1u | (1u << 30) | (1u << 31);
  lg0[1] = lds_off;
  lg0[2] = (uint32_t)(wAddr & 0xFFFFFFFFull);
  lg0[3] = (uint32_t)((wAddr >> 32) & 0x01FFFFFFull) | (2u << 30);

  // group1: data_size=2 (4B); tensor_dim0=D; tensor_dim1=VOCAB;
  //         tile_dim0=D; tile_dim1=#indices=8; tensor_dim0_stride=D
  i32x8 lg1;
  lg1[0] = (int)(2u << 16);                                         // data_size
  lg1[1] = (int)((EMB_D & 0xFFFFu) << 16);                          // dim0[15:0]
  lg1[2] = (int)(((EMB_D >> 16) & 0xFFFFu)
               | ((EMB_VOCAB & 0xFFFFu) << 16));                    // dim0hi|dim1lo
  lg1[3] = (int)((((EMB_VOCAB >> 16) & 0xFFFFu))
               | ((EMB_D & 0xFFFFu) << 16));                        // dim1hi|tile_dim0
  lg1[4] = (int)ROWS_PER_WG;                                        // tile_dim1 (=#idx)
  lg1[5] = (int)EMB_D;                                              // dim0_stride[31:0]
  lg1[6] = 0;
  lg1[7] = 0;

  // groups 2/3: eight 32-bit row indices
  i32x4 lg2; lg2[0] = i0; lg2[1] = i1; lg2[2] = i2; lg2[3] = i3;
  i32x4 lg3; lg3[0] = i4; lg3[1] = i5; lg3[2] = i6; lg3[3] = i7;

  TDM_LOAD(lg0, lg1, lg2, lg3);

  // Tile must be resident in LDS before the store descriptor reads it.
  __builtin_amdgcn_s_wait_tensorcnt(0);

  // ---------------- store descriptor: normal mode, contiguous 2-D ---------
  u32x4 sg0;
  sg0[0] = 1u;                                                      // count=1
  sg0[1] = lds_off;
  sg0[2] = (uint32_t)(oAddr & 0xFFFFFFFFull);
  sg0[3] = (uint32_t)((oAddr >> 32) & 0x01FFFFFFull) | (2u << 30);

  i32x8 sg1;
  sg1[0] = (int)(2u << 16);                                         // data_size=4B
  sg1[1] = (int)((EMB_D & 0xFFFFu) << 16);                          // dim0[15:0]
  sg1[2] = (int)(((EMB_D >> 16) & 0xFFFFu)
               | ((ROWS_PER_WG & 0xFFFFu) << 16));                  // dim1 = 8
  sg1[3] = (int)(((ROWS_PER_WG >> 16) & 0xFFFFu)
               | ((EMB_D & 0xFFFFu) << 16));                        // tile_dim0 = D
  sg1[4] = (int)ROWS_PER_WG;                                        // tile_dim1 = 8
  sg1[5] = (int)EMB_D;                                              // dim0_stride
  sg1[6] = 0;
  sg1[7] = 0;

  i32x4 z4; z4[0] = 0; z4[1] = 0; z4[2] = 0; z4[3] = 0;             // dims 2+ unused

  TDM_STORE(sg0, sg1, z4, z4);
  // S_ENDPGM performs an implicit wait-idle (TENSORcnt drains before exit).
}

extern "C" void kernel_launch(void* const* d_in, const int* in_sizes, int n_in,
                              void* d_out, int out_size, void* d_ws, size_t ws_size,
                              hipStream_t stream) {
  const int*   x   = (const int*)d_in[0];    // (B*S) int32 token ids
  const float* W   = (const float*)d_in[1];  // (VOCAB, D) float32
  float*       out = (float*)d_out;          // (B*S, D) float32

  const int n_tokens = in_sizes[0];                       // 16384
  const int n_blocks = n_tokens / (int)ROWS_PER_WG;       // 2048 WGs, 1 wave each

  embed_gather_tdm<<<n_blocks, 32, 0, stream>>>(x, W, out);
}